// FrozenSentenceEncoder_78658031059404
// MI455X (gfx1250) — compile-verified
//
#include <hip/hip_runtime.h>
#include <hip/hip_bf16.h>
#include <stdint.h>

#define DIM      768
#define SEQ      512
#define TPB      256
#define RPB      8          // rows per block, double-buffered async pipeline
#define NWAVES   (TPB / 32)

static constexpr int MULT_MOD = 1315423911 % DIM;   // == 423

// Truncate a generic (flat) pointer to a __shared__ object down to the 32-bit
// LDS offset expected by async-to-LDS vdst (flat shared addr = aperture | lds_off).
__device__ __forceinline__ unsigned lds_off(const void* p) {
    return (unsigned)(uintptr_t)p;
}

__global__ __launch_bounds__(TPB) void
FrozenSentenceEncoder_hist_kernel(const int* __restrict__ codes,
                                  const float* __restrict__ wts,
                                  float* __restrict__ out, int B)
{
    __shared__ int   sh_codes[2][SEQ];   // 4 KB
    __shared__ float sh_w[2][SEQ];       // 4 KB
    __shared__ float hist[DIM];          // 3 KB
    __shared__ float red[NWAVES];

    const int t    = threadIdx.x;
    const int wave = t >> 5;
    const int row0 = blockIdx.x * RPB;

    // Issue per-lane async copies: 256 lanes x 8B = 2048 B (one row of codes),
    // same for weights. Data lands in LDS without touching VGPRs.
    auto prefetch = [&](int row, int buf) {
        int r = row < B ? row : B - 1;               // clamp: stay in-bounds, data unused
        uint64_t gc = (uint64_t)(uintptr_t)(codes + (size_t)r * SEQ + 2 * t);
        uint64_t gw = (uint64_t)(uintptr_t)(wts   + (size_t)r * SEQ + 2 * t);
        unsigned lc = lds_off(&sh_codes[buf][2 * t]);
        unsigned lw = lds_off(&sh_w[buf][2 * t]);
        asm volatile("global_load_async_to_lds_b64 %0, %1, off"
                     :: "v"(lc), "v"(gc) : "memory");
        asm volatile("global_load_async_to_lds_b64 %0, %1, off"
                     :: "v"(lw), "v"(gw) : "memory");
    };

    prefetch(row0, 0);
    int buf = 0;

    for (int i = 0; i < RPB; ++i) {
        const int row = row0 + i;                    // uniform across the block

        if (i + 1 < RPB) {
            prefetch(row0 + i + 1, buf ^ 1);
            // 4 outstanding per wave; <=2 left means current buffer's loads done
            asm volatile("s_wait_asynccnt 2" ::: "memory");
        } else {
            asm volatile("s_wait_asynccnt 0" ::: "memory");
        }

        // Zero the histogram while waiting on other waves
        #pragma unroll
        for (int k = 0; k < DIM / TPB; ++k)
            hist[t + k * TPB] = 0.0f;
        __syncthreads();   // async data visible block-wide, hist zeroed

        // Scatter: 2 chars per thread -> ds_add_f32 LDS atomics
        #pragma unroll
        for (int k = 0; k < SEQ / TPB; ++k) {
            int   c = sh_codes[buf][t + k * TPB];
            float w = sh_w[buf][t + k * TPB];
            int bin = ((c % DIM) * MULT_MOD) % DIM;  // exact reference hash
            atomicAdd(&hist[bin], w);
        }
        __syncthreads();

        // Sum of squares: 3 bins/thread, wave32 shuffle tree, cross-wave via LDS
        float s = 0.0f;
        #pragma unroll
        for (int k = 0; k < DIM / TPB; ++k) {
            float v = hist[t + k * TPB];
            s += v * v;
        }
        #pragma unroll
        for (int off = 16; off > 0; off >>= 1)
            s += __shfl_xor(s, off, 32);
        if ((t & 31) == 0) red[wave] = s;
        __syncthreads();

        float tot = 0.0f;
        #pragma unroll
        for (int wv = 0; wv < NWAVES; ++wv) tot += red[wv];
        float scale = 1.0f / (sqrtf(tot) + 1e-6f);

        if (row < B) {
            float* orow = out + (size_t)row * DIM;
            #pragma unroll
            for (int k = 0; k < DIM / TPB; ++k) {
                int idx = t + k * TPB;
                // output is write-once: bypass L2 residency with NT stores
                __builtin_nontemporal_store(hist[idx] * scale, orow + idx);
            }
        }
        __syncthreads();   // all hist readers done before next iter re-zeroes
        buf ^= 1;
    }
}

extern "C" void kernel_launch(void* const* d_in, const int* in_sizes, int n_in,
                              void* d_out, int out_size, void* d_ws, size_t ws_size,
                              hipStream_t stream) {
    const int*   codes = (const int*)d_in[0];
    const float* wts   = (const float*)d_in[1];
    float*       out   = (float*)d_out;

    const int B    = in_sizes[0] / SEQ;              // 32768
    const int grid = (B + RPB - 1) / RPB;            // 4096 blocks

    FrozenSentenceEncoder_hist_kernel<<<grid, TPB, 0, stream>>>(codes, wts, out, B);
}